// APPNProp_1580547966593
// MI455X (gfx1250) — compile-verified
//
#include <hip/hip_runtime.h>

// ---------------------------------------------------------------------------
// APPNP propagation on MI455X (gfx1250, wave32).
//
//   h_{t+1} = 0.9 * A h_t + 0.1 * x,  10 steps.
//   A: 100k x 100k COO, 1.6M nnz.  h: 100000 x 128 fp32.
//
// Roofline: SpMM step = 0.41 GFLOP vs ~1.6 GB of gather/scatter traffic
// (0.25 FLOP/B) -> pure data movement. Key MI455X lever: h (51.2MB) +
// accumulator (51.2MB) + packed edges (19.2MB) all fit in the 192MB L2,
// so the random gather/scatter-atomic stream runs at L2 speed.
//
// WMMA: the dominant scatter phase cannot use matrix units, so the
// "y = alpha*x" init pass (bandwidth-trivial) is done with
// V_WMMA_F32_16X16X4_F32 as  D = X_tile * (0.1 * I16)  (4 chained K=4
// WMMAs per 16x16 tile) -- numerically identical to a scalar scale and
// free, since that pass is memory bound regardless.
// ---------------------------------------------------------------------------

#define N_NODES 100000
#define D_FEAT  128
#define N_EDGES 1600000
#define ALPHA   0.1f
#define ONE_MINUS_ALPHA 0.9f

typedef float v2f __attribute__((ext_vector_type(2)));
typedef float v8f __attribute__((ext_vector_type(8)));

// --------------------------------------------------------------------------
// Pass 0 (once per launch): int64 COO -> int32 rows/cols + prescaled vals.
// --------------------------------------------------------------------------
__global__ void pack_edges_kernel(const long long* __restrict__ edge_index,
                                  const float* __restrict__ edge_vals,
                                  int* __restrict__ erow,
                                  int* __restrict__ ecol,
                                  float* __restrict__ evals) {
    int e = blockIdx.x * blockDim.x + threadIdx.x;
    if (e < N_EDGES) {
        erow[e]  = (int)edge_index[e];            // row = edge_index[0][e]
        ecol[e]  = (int)edge_index[N_EDGES + e];  // col = edge_index[1][e]
        evals[e] = ONE_MINUS_ALPHA * edge_vals[e];
    }
}

// --------------------------------------------------------------------------
// Per-step init: y = 0.1 * x, via V_WMMA_F32_16X16X4_F32.
// One wave per 16x16 tile (16 nodes x 16 features).
//   D(16x16) = sum_t A_t(16x4) * B_t(4x16),  A_t = x[:, 4t:4t+4] chunk,
//   B_t = 0.1 * I16[4t:4t+4, :]   =>   D = 0.1 * x tile, exactly.
//
// f32 A 16x4 lane layout (ISA 7.12.2): lane = M (mod 16);
//   VGPR0 holds K=0 (lanes 0-15) / K=2 (lanes 16-31), VGPR1 holds K=1 / K=3.
// B 4x16 mirrors with N = lane&15. D: VGPR r -> M=r (lanes 0-15), M=r+8.
// Grid is exact (no guard): EXEC all-1s as WMMA requires.
// --------------------------------------------------------------------------
__global__ void init_alpha_x_wmma(const float* __restrict__ x,
                                  float* __restrict__ y) {
    const int wave  = (blockIdx.x * blockDim.x + threadIdx.x) >> 5;
    const int lane  = threadIdx.x & 31;
    const int ftile = wave & 7;          // 128/16 = 8 feature tiles
    const int ntile = wave >> 3;         // 100000/16 = 6250 node tiles
    const int node0 = ntile << 4;
    const int f0    = ftile << 4;

    const int m    = lane & 15;          // A row / half-select
    const int n    = lane & 15;          // B/D column
    const int koff = (lane < 16) ? 0 : 2;

    const float* __restrict__ xrow = x + (size_t)(node0 + m) * D_FEAT + f0;

    v8f acc = {0.f, 0.f, 0.f, 0.f, 0.f, 0.f, 0.f, 0.f};
#pragma unroll
    for (int t = 0; t < 4; ++t) {
        v2f a;
        a.x = xrow[4 * t + koff];        // A[m, 4t+koff]
        a.y = xrow[4 * t + koff + 1];    // A[m, 4t+koff+1]
        v2f b;                           // B = 0.1 * I rows 4t..4t+3
        b.x = (n == 4 * t + koff)     ? ALPHA : 0.0f;
        b.y = (n == 4 * t + koff + 1) ? ALPHA : 0.0f;
        acc = __builtin_amdgcn_wmma_f32_16x16x4_f32(
            /*neg_a=*/false, a, /*neg_b=*/false, b,
            /*c_mod=*/(short)0, acc, /*reuse_a=*/false, /*reuse_b=*/false);
    }

    const int mbase = (lane < 16) ? 0 : 8;
#pragma unroll
    for (int r = 0; r < 8; ++r) {
        y[(size_t)(node0 + mbase + r) * D_FEAT + f0 + n] = acc[r];
    }
}

// --------------------------------------------------------------------------
// Per-step SpMM scatter: one wave32 per edge.
// 32 lanes x float4 = the full 128-feature row (512B coalesced gather),
// then 4 global_atomic_add_f32 per lane into the destination row.
// All buffers are L2-resident -> runs at L2 gather/atomic throughput.
// --------------------------------------------------------------------------
__global__ void spmm_scatter_kernel(const int* __restrict__ erow,
                                    const int* __restrict__ ecol,
                                    const float* __restrict__ evals,
                                    const float* __restrict__ hin,
                                    float* __restrict__ yout) {
    const int e    = (blockIdx.x * blockDim.x + threadIdx.x) >> 5;
    const int lane = threadIdx.x & 31;

    const int   row = erow[e];      // uniform within wave
    const int   col = ecol[e];
    const float v   = evals[e];     // already scaled by (1 - alpha)

    const float4* __restrict__ src =
        (const float4*)(hin + (size_t)col * D_FEAT);
    const float4 msg = src[lane];

    float* dst = yout + (size_t)row * D_FEAT + (lane << 2);
    atomicAdd(dst + 0, v * msg.x);
    atomicAdd(dst + 1, v * msg.y);
    atomicAdd(dst + 2, v * msg.z);
    atomicAdd(dst + 3, v * msg.w);
}

// --------------------------------------------------------------------------
// Launch: pack once, then 10 x (init, scatter), ping-pong ws <-> d_out.
// ws layout: [ bufA: N*D floats | erow: E ints | ecol: E ints | evals: E floats ]
//            = 51.2MB + 19.2MB = 70.4MB.
// Step s writes bufA for even s, d_out for odd s -> step 9 lands in d_out.
// --------------------------------------------------------------------------
extern "C" void kernel_launch(void* const* d_in, const int* in_sizes, int n_in,
                              void* d_out, int out_size, void* d_ws, size_t ws_size,
                              hipStream_t stream) {
    const float*     x  = (const float*)d_in[0];
    const long long* ei = (const long long*)d_in[1];   // int64 edge_index [2, E]
    const float*     ev = (const float*)d_in[2];
    float*           out = (float*)d_out;

    float* bufA  = (float*)d_ws;
    int*   erow  = (int*)(bufA + (size_t)N_NODES * D_FEAT);
    int*   ecol  = erow + N_EDGES;
    float* evals = (float*)(ecol + N_EDGES);

    pack_edges_kernel<<<(N_EDGES + 255) / 256, 256, 0, stream>>>(ei, ev, erow,
                                                                 ecol, evals);

    // init grid: (100000/16)*(128/16) = 50000 waves = 1,600,000 threads (exact)
    const int init_blocks = (N_NODES / 16) * (D_FEAT / 16) * 32 / 256;  // 6250
    // spmm grid: one wave per edge, 8 edges per 256-thread block
    const int spmm_blocks = N_EDGES / 8;                                // 200000

    const float* hin = x;
    for (int s = 0; s < 10; ++s) {
        float* hout = (s & 1) ? out : bufA;
        init_alpha_x_wmma<<<init_blocks, 256, 0, stream>>>(x, hout);
        spmm_scatter_kernel<<<spmm_blocks, 256, 0, stream>>>(erow, ecol, evals,
                                                             hin, hout);
        hin = hout;
    }
}